// BondingGraphGNN_6339371729325
// MI455X (gfx1250) — compile-verified
//
#include <hip/hip_runtime.h>
#include <hip/hip_bf16.h>

#define F_IN 92
#define HDIM 128

typedef float v2f __attribute__((ext_vector_type(2)));
typedef float v8f __attribute__((ext_vector_type(8)));

__device__ __forceinline__ float sigmoidf_(float x) {
    return 1.0f / (1.0f + __expf(-x));
}

// ---------------------------------------------------------------------------
// C[M x 128] = op(A[M x K(lda)] @ B[K x 128]); one wave -> one 16x16 tile.
// f32 WMMA 16x16x4: A frag lanes0-15 K=0,1 ; lanes16-31 K=2,3 (v2f)
//                   B frag VGPR0: K = 0 / 2 per lane half; VGPR1: K = 1 / 3
//                   C/D:   VGPR v: row = v (lanes0-15) / v+8 (lanes16-31)
// ---------------------------------------------------------------------------
template <bool TANH>
__global__ __launch_bounds__(256) void gemm_nn_kernel(const float* __restrict__ A,
                                                      const float* __restrict__ B,
                                                      float* __restrict__ C,
                                                      int K, int lda, int N) {
    const int wave = threadIdx.x >> 5;
    const int lane = threadIdx.x & 31;
    const int half = lane >> 4;
    const int r    = lane & 15;
    const int m0   = blockIdx.x * 16;
    const int n0   = wave * 16;

    int arow_i = m0 + r;
    if (arow_i >= N) arow_i = 0;                 // clamp (no divergence: EXEC stays full)
    const float* __restrict__ arow = A + (size_t)arow_i * lda;
    const float* __restrict__ bcol = B + n0 + r;

    v8f c = {};
    for (int kk = 0; kk < K; kk += 4) {
        const int ka = kk + 2 * half;
        v2f a; a.x = arow[ka];                 a.y = arow[ka + 1];
        v2f b; b.x = bcol[(size_t)ka * HDIM];  b.y = bcol[(size_t)(ka + 1) * HDIM];
        c = __builtin_amdgcn_wmma_f32_16x16x4_f32(false, a, false, b,
                                                  (short)0, c, false, false);
    }
    #pragma unroll
    for (int v = 0; v < 8; ++v) {
        const int row = m0 + v + 8 * half;
        if (row < N) {
            float val = c[v];
            if (TANH) val = tanhf(val);
            C[(size_t)row * HDIM + n0 + r] = val;
        }
    }
}

// ---------------------------------------------------------------------------
// Edge scatter: agg[dst,:] += m[src,:]. One wave per edge, float4 per lane.
// ---------------------------------------------------------------------------
__global__ __launch_bounds__(256) void scatter_kernel(const float* __restrict__ m,
                                                      const int* __restrict__ src,
                                                      const int* __restrict__ dst,
                                                      float* __restrict__ agg, int E) {
    const int e = (int)((blockIdx.x * (size_t)blockDim.x + threadIdx.x) >> 5);
    if (e >= E) return;
    const int lane = threadIdx.x & 31;
    const int s = src[e];
    const int d = dst[e];
    const float4 v = *((const float4*)(m + (size_t)s * HDIM) + lane);
    float* ap = agg + (size_t)d * HDIM + lane * 4;
    atomicAdd(ap + 0, v.x);
    atomicAdd(ap + 1, v.y);
    atomicAdd(ap + 2, v.z);
    atomicAdd(ap + 3, v.w);
}

// ---------------------------------------------------------------------------
// Fused GRUCell: gi = agg @ w_ih^T (+b_ih), gh = h @ w_hh^T (+b_hh), gates,
// h_new -> hOut. One wave computes one 16x16 tile of h_new; the six needed
// GEMM tiles (K=128) are accumulated sharing A fragments across 3 WMMAs each.
// w_ih/w_hh are [3H x H] row-major => B^T loads are contiguous per lane.
// ---------------------------------------------------------------------------
__global__ __launch_bounds__(256) void gru_kernel(const float* __restrict__ agg,
                                                  const float* __restrict__ hIn,
                                                  const float* __restrict__ w_ih,
                                                  const float* __restrict__ w_hh,
                                                  const float* __restrict__ b_ih,
                                                  const float* __restrict__ b_hh,
                                                  float* __restrict__ hOut, int N) {
    const int wave = threadIdx.x >> 5;
    const int lane = threadIdx.x & 31;
    const int half = lane >> 4;
    const int r    = lane & 15;
    const int m0   = blockIdx.x * 16;
    const int n0   = wave * 16;

    int arow_i = m0 + r;
    if (arow_i >= N) arow_i = 0;
    const float* __restrict__ aggrow = agg + (size_t)arow_i * HDIM;
    const float* __restrict__ hrow   = hIn + (size_t)arow_i * HDIM;

    const int col = n0 + r;
    const float* __restrict__ wir = w_ih + (size_t)(col)            * HDIM;
    const float* __restrict__ wiz = w_ih + (size_t)(col + HDIM)     * HDIM;
    const float* __restrict__ win = w_ih + (size_t)(col + 2 * HDIM) * HDIM;
    const float* __restrict__ whr = w_hh + (size_t)(col)            * HDIM;
    const float* __restrict__ whz = w_hh + (size_t)(col + HDIM)     * HDIM;
    const float* __restrict__ whn = w_hh + (size_t)(col + 2 * HDIM) * HDIM;

    v8f cir = {}, ciz = {}, cin = {}, chr_ = {}, chz = {}, chn = {};
    for (int kk = 0; kk < HDIM; kk += 4) {
        const int ka = kk + 2 * half;
        v2f aa; aa.x = aggrow[ka]; aa.y = aggrow[ka + 1];
        v2f ah; ah.x = hrow[ka];   ah.y = hrow[ka + 1];
        v2f b;
        b.x = wir[ka]; b.y = wir[ka + 1];
        cir = __builtin_amdgcn_wmma_f32_16x16x4_f32(false, aa, false, b, (short)0, cir, false, false);
        b.x = wiz[ka]; b.y = wiz[ka + 1];
        ciz = __builtin_amdgcn_wmma_f32_16x16x4_f32(false, aa, false, b, (short)0, ciz, false, false);
        b.x = win[ka]; b.y = win[ka + 1];
        cin = __builtin_amdgcn_wmma_f32_16x16x4_f32(false, aa, false, b, (short)0, cin, false, false);
        b.x = whr[ka]; b.y = whr[ka + 1];
        chr_ = __builtin_amdgcn_wmma_f32_16x16x4_f32(false, ah, false, b, (short)0, chr_, false, false);
        b.x = whz[ka]; b.y = whz[ka + 1];
        chz = __builtin_amdgcn_wmma_f32_16x16x4_f32(false, ah, false, b, (short)0, chz, false, false);
        b.x = whn[ka]; b.y = whn[ka + 1];
        chn = __builtin_amdgcn_wmma_f32_16x16x4_f32(false, ah, false, b, (short)0, chn, false, false);
    }

    const float bir = b_ih[col], biz = b_ih[col + HDIM], bin = b_ih[col + 2 * HDIM];
    const float bhr = b_hh[col], bhz = b_hh[col + HDIM], bhn = b_hh[col + 2 * HDIM];

    #pragma unroll
    for (int v = 0; v < 8; ++v) {
        const int row = m0 + v + 8 * half;
        if (row < N) {
            const float ho = hIn[(size_t)row * HDIM + col];
            const float rg = sigmoidf_(cir[v] + bir + chr_[v] + bhr);
            const float zg = sigmoidf_(ciz[v] + biz + chz[v] + bhz);
            const float cd = tanhf(cin[v] + bin + rg * (chn[v] + bhn));
            hOut[(size_t)row * HDIM + col] = (1.0f - zg) * cd + zg * ho;
        }
    }
}

// ---------------------------------------------------------------------------
// Global mean-pool accumulation (batch is sorted -> good atomic locality).
// ---------------------------------------------------------------------------
__global__ __launch_bounds__(256) void pool_kernel(const float* __restrict__ h,
                                                   const int* __restrict__ batch,
                                                   float* __restrict__ sums,
                                                   float* __restrict__ counts, int N) {
    const size_t idx = blockIdx.x * (size_t)blockDim.x + threadIdx.x;
    if (idx >= (size_t)N * HDIM) return;
    const int i = (int)(idx >> 7);
    const int c = (int)(idx & (HDIM - 1));
    const int b = batch[i];
    atomicAdd(&sums[(size_t)b * HDIM + c], h[idx]);
    if (c == 0) atomicAdd(&counts[b], 1.0f);
}

// ---------------------------------------------------------------------------
// out[g] = relu(sums[g,:] / max(count,1)) . w_pred + b_pred ; one wave/graph.
// ---------------------------------------------------------------------------
__global__ __launch_bounds__(32) void pred_kernel(const float* __restrict__ sums,
                                                  const float* __restrict__ counts,
                                                  const float* __restrict__ w_pred,
                                                  const float* __restrict__ b_pred,
                                                  float* __restrict__ out) {
    const int g = blockIdx.x;
    const int lane = threadIdx.x;
    const float inv = 1.0f / fmaxf(counts[g], 1.0f);
    float acc = 0.0f;
    #pragma unroll
    for (int t = 0; t < 4; ++t) {
        const int c = lane * 4 + t;
        const float v = fmaxf(sums[(size_t)g * HDIM + c] * inv, 0.0f);
        acc += v * w_pred[c];
    }
    #pragma unroll
    for (int off = 16; off > 0; off >>= 1) acc += __shfl_xor(acc, off, 32);
    if (lane == 0) out[g] = acc + b_pred[0];
}

extern "C" void kernel_launch(void* const* d_in, const int* in_sizes, int n_in,
                              void* d_out, int out_size, void* d_ws, size_t ws_size,
                              hipStream_t stream) {
    const float* x       = (const float*)d_in[0];
    const int*   ei      = (const int*)d_in[1];   // [2, E] flat: src then dst
    const int*   batch   = (const int*)d_in[2];
    const float* W_embed = (const float*)d_in[3]; // [F, H]
    const float* W_mpnn  = (const float*)d_in[4]; // [6, H, H]
    const float* w_ih    = (const float*)d_in[5]; // [3H, H]
    const float* w_hh    = (const float*)d_in[6]; // [3H, H]
    const float* b_ih    = (const float*)d_in[7];
    const float* b_hh    = (const float*)d_in[8];
    const float* w_pred  = (const float*)d_in[9];
    const float* b_pred  = (const float*)d_in[10];
    float* out = (float*)d_out;

    const int N = in_sizes[0] / F_IN;
    const int E = in_sizes[1] / 2;
    const int G = out_size;
    const size_t NH = (size_t)N * HDIM;

    float* bufA   = (float*)d_ws;     // h ping
    float* bufB   = bufA + NH;        // m / h pong
    float* agg    = bufB + NH;
    float* sums   = agg + NH;
    float* counts = sums + (size_t)G * HDIM;

    const int mtiles = (N + 15) / 16;

    // h = tanh(x @ W_embed)
    gemm_nn_kernel<true><<<mtiles, 256, 0, stream>>>(x, W_embed, bufA, F_IN, F_IN, N);

    float* hIn = bufA;
    float* hOut = bufB;
    for (int s = 0; s < 6; ++s) {
        // m = h @ W_mpnn[s]  (into the pong buffer)
        gemm_nn_kernel<false><<<mtiles, 256, 0, stream>>>(
            hIn, W_mpnn + (size_t)s * HDIM * HDIM, hOut, HDIM, HDIM, N);
        // agg = scatter_add(m[src], dst)
        hipMemsetAsync(agg, 0, NH * sizeof(float), stream);
        const int sblocks = (int)(((size_t)E * 32 + 255) / 256);
        scatter_kernel<<<sblocks, 256, 0, stream>>>(hOut, ei, ei + E, agg, E);
        // h_new = GRU(agg, h) -> overwrites dead message buffer (ping-pong)
        gru_kernel<<<mtiles, 256, 0, stream>>>(agg, hIn, w_ih, w_hh, b_ih, b_hh, hOut, N);
        float* t = hIn; hIn = hOut; hOut = t;
    }

    hipMemsetAsync(sums, 0, (size_t)G * HDIM * sizeof(float), stream);
    hipMemsetAsync(counts, 0, (size_t)G * sizeof(float), stream);
    const int pblocks = (int)((NH + 255) / 256);
    pool_kernel<<<pblocks, 256, 0, stream>>>(hIn, batch, sums, counts, N);
    pred_kernel<<<G, 32, 0, stream>>>(sums, counts, w_pred, b_pred, out);
}